// HeterogeneousAttentionDistillationLoss_68891275428342
// MI455X (gfx1250) — compile-verified
//
#include <hip/hip_runtime.h>
#include <math.h>

// ---------------------------------------------------------------------------
// HeterogeneousAttentionDistillationLoss for MI455X (gfx1250, wave32).
// HBM-bandwidth-bound loss (~245MB moved => ~10us floor @ 23.3TB/s).
//   - all streaming in b128 (float4) loads/stores, 512B/wave/request
//   - read-twice tensors (student/teacher) left RT to hit 192MB L2 on pass 2;
//     read-once tensors loaded non-temporal (TH=NT)
//   - V_WMMA_F32_16X16X4_F32 (ones-matrix trick) for all block reductions
//   - deterministic: no float atomics; fixed-order partial sums in ws
// ---------------------------------------------------------------------------

typedef __attribute__((ext_vector_type(2))) float v2f;
typedef __attribute__((ext_vector_type(4))) float v4f;
typedef __attribute__((ext_vector_type(8))) float v8f;

#if __has_builtin(__builtin_amdgcn_wmma_f32_16x16x4_f32)
#define CDNA5_USE_WMMA 1
#else
#define CDNA5_USE_WMMA 0
#endif

#define B_     8
#define C_     256
#define HW_    4096        // 64*64
#define BC_    2048        // B*C
#define BHW_   32768       // B*H*W
#define HO2_   65536       // 256*256
#define FEAT_N 8388608.0f  // B*C*H*W
#define KL_BLOCKS 512

// workspace layout (float offsets) — everything written before read each call
#define WS_CHAN_S   0
#define WS_CHAN_T   2048
#define WS_PART_G   4096
#define WS_PART_L   6144
#define WS_PART_KL  8192
#define WS_PART_C1  8704
#define WS_OD       9216        // /4 = 2304  (v4f index)
#define WS_HD       41984       // /4 = 10496
#define WS_SSP      74752       // /4 = 18688
#define WS_TSP      107520      // /4 = 26880
#define WS_PART_SP  140288      // /4 = 35072; 4 chunks * 8 quantities * 32768
// total: 140288 + 1048576 floats = 4.54 MB of ws

__device__ __forceinline__ float sigm(float x) {
  return 1.0f / (1.0f + __expf(-x));
}
__device__ __forceinline__ v4f v4max(v4f a, v4f b) {
  v4f r; r.x = fmaxf(a.x, b.x); r.y = fmaxf(a.y, b.y);
  r.z = fmaxf(a.z, b.z); r.w = fmaxf(a.w, b.w); return r;
}
__device__ __forceinline__ v4f v4abs(v4f a) {
  v4f r; r.x = fabsf(a.x); r.y = fabsf(a.y);
  r.z = fabsf(a.z); r.w = fabsf(a.w); return r;
}

// Deterministic block-wide sum of 256 per-thread values.
// Final 64->1 stage: one V_WMMA_F32_16X16X4_F32 with B = ones:
//   D[m,n] = rowsum_A(m); only the documented C/D layout is relied upon
//   (lane L, vgpr v -> D[v + 8*(L/16), L%16]): per-lane sum of the 8 D vgprs
//   covers half the rows, xor-16 pairs the halves => exact total in all lanes.
__device__ __forceinline__ float block_sum_256(float v, float* sm) {
  const int tid = threadIdx.x;
  sm[tid] = v;
  __syncthreads();
  if (tid < 64) sm[tid] += sm[tid + 64] + sm[tid + 128] + sm[tid + 192];
  __syncthreads();
  float tot = 0.0f;
  if (tid < 32) {  // wave 0, EXEC all-ones (WMMA requirement)
#if CDNA5_USE_WMMA
    v2f a;  a.x = sm[2 * tid];  a.y = sm[2 * tid + 1];   // 64 partials -> A
    v2f ones; ones.x = 1.0f; ones.y = 1.0f;              // B = ones(4x16)
    v8f c = {};
    v8f d = __builtin_amdgcn_wmma_f32_16x16x4_f32(
        false, a, false, ones, (short)0, c, false, false);
    float s = d[0] + d[1] + d[2] + d[3] + d[4] + d[5] + d[6] + d[7];
    s += __shfl_xor(s, 16, 32);
    tot = s;
#else
    float s = sm[tid] + sm[tid + 32];
    s += __shfl_xor(s, 16, 32);
    s += __shfl_xor(s, 8, 32);
    s += __shfl_xor(s, 4, 32);
    s += __shfl_xor(s, 2, 32);
    s += __shfl_xor(s, 1, 32);
    tot = s;
#endif
  }
  __syncthreads();
  if (tid == 0) sm[0] = tot;
  __syncthreads();
  float r = sm[0];
  __syncthreads();
  return r;
}

// ---- Pass 1: channel-major over student/teacher (b128 streaming) ------------
// One block per (b,c): contiguous 16KB rows. Per-channel sums + per-block
// partials of global/local squared error.
__global__ void __launch_bounds__(256)
feat_channel_kernel(const float* __restrict__ S, const float* __restrict__ T,
                    const float* __restrict__ M, float* __restrict__ ws) {
  __shared__ float sm[256];
  const int bc = blockIdx.x;
  const int b  = bc >> 8;
  const v4f* s4 = (const v4f*)(S + (size_t)bc * HW_);
  const v4f* t4 = (const v4f*)(T + (size_t)bc * HW_);
  const v4f* m4 = (const v4f*)(M + (size_t)b * HW_);
  v4f g4 = {0,0,0,0}, l4 = g4, ss4 = g4, st4 = g4;
  for (int i = threadIdx.x; i < 1024; i += 256) {  // 1024 v4f per channel
    v4f sv = s4[i], tv = t4[i], mv = m4[i];
    v4f d = sv - tv;
    g4 += d * d;
    v4f md = mv * d;            // mse(s*m, t*m) = mean(m^2 (s-t)^2)
    l4 += md * md;
    ss4 += sv;  st4 += tv;
  }
  float gs  = block_sum_256(g4.x + g4.y + g4.z + g4.w, sm);
  float ls  = block_sum_256(l4.x + l4.y + l4.z + l4.w, sm);
  float sss = block_sum_256(ss4.x + ss4.y + ss4.z + ss4.w, sm);
  float sts = block_sum_256(st4.x + st4.y + st4.z + st4.w, sm);
  if (threadIdx.x == 0) {
    ws[WS_PART_G + bc] = gs;
    ws[WS_PART_L + bc] = ls;
    ws[WS_CHAN_S + bc] = sss;
    ws[WS_CHAN_T + bc] = sts;
  }
}

// ---- Pass 2a: position-major fused sweep, C split into 4 chunks -------------
// 128 blocks = 8 b * 4 pixel-tiles * 4 c-chunks. Each lane owns 4 consecutive
// pixels (v4f): 512B contiguous per wave per stream per iteration.
// opt_t1/opt_t2/sar_t2 are read-once -> non-temporal; S/T reused -> RT + prefetch.
__global__ void __launch_bounds__(256)
spatial_diff_partial_kernel(const float* __restrict__ S,  const float* __restrict__ T,
                            const float* __restrict__ O1, const float* __restrict__ O2,
                            const float* __restrict__ SR, float* __restrict__ ws) {
  const int blk  = blockIdx.x;
  const int b    = blk >> 4;
  const int tile = (blk >> 2) & 3;
  const int k    = blk & 3;                       // c-chunk: [k*64, k*64+64)
  const int p4   = (tile << 8) + threadIdx.x;     // v4f pixel in [0,1024)
  const v4f* S4  = (const v4f*)S;
  const v4f* T4  = (const v4f*)T;
  const v4f* O14 = (const v4f*)O1;
  const v4f* O24 = (const v4f*)O2;
  const v4f* SR4 = (const v4f*)SR;
  const size_t base = ((size_t)b * C_ + (size_t)k * 64) * 1024 + (size_t)p4;

  v4f ss = {0,0,0,0}, ts = ss;
  v4f sx = {-3.4e38f,-3.4e38f,-3.4e38f,-3.4e38f}, tx = sx;
  v4f lo = ss, co = ss, lh = ss, ch = ss;
  const v4f eps = {1e-8f,1e-8f,1e-8f,1e-8f};
  for (int c = 0; c < 64; ++c) {
    size_t idx = base + (size_t)c * 1024;
    if (c + 8 < 64) {
      __builtin_prefetch(&S4[idx + 8 * 1024], 0, 0);   // -> global_prefetch_b8
      __builtin_prefetch(&T4[idx + 8 * 1024], 0, 0);
    }
    v4f sv = S4[idx], tv = T4[idx];
    v4f o1 = __builtin_nontemporal_load(&O14[idx]);
    v4f o2 = __builtin_nontemporal_load(&O24[idx]);
    v4f sr = __builtin_nontemporal_load(&SR4[idx]);
    ss += sv;  ts += tv;
    sx = v4max(sx, sv);  tx = v4max(tx, tv);
    v4f d  = o1 - o2;  lo += d * d;
    v4f p12 = o1 * o2; co += p12 / v4max(v4abs(p12), eps);
    v4f dh = o1 - sr;  lh += dh * dh;
    v4f p1s = o1 * sr; ch += p1s / v4max(v4abs(p1s), eps);
  }
  v4f* wsv = (v4f*)ws;
  const int gp4 = b * 1024 + p4;                  // global v4f pixel [0,8192)
  const int PSV = WS_PART_SP / 4;
  wsv[PSV + (k * 8 + 0) * 8192 + gp4] = ss;
  wsv[PSV + (k * 8 + 1) * 8192 + gp4] = ts;
  wsv[PSV + (k * 8 + 2) * 8192 + gp4] = sx;
  wsv[PSV + (k * 8 + 3) * 8192 + gp4] = tx;
  wsv[PSV + (k * 8 + 4) * 8192 + gp4] = lo;
  wsv[PSV + (k * 8 + 5) * 8192 + gp4] = co;
  wsv[PSV + (k * 8 + 6) * 8192 + gp4] = lh;
  wsv[PSV + (k * 8 + 7) * 8192 + gp4] = ch;
}

// ---- Pass 2b: combine c-chunks, produce attention + difference maps ---------
__global__ void __launch_bounds__(256)
spatial_diff_combine_kernel(float* __restrict__ ws) {
  const int gp4 = blockIdx.x * 256 + threadIdx.x;  // [0,8192)
  const v4f* wsv = (const v4f*)ws;
  v4f* wso = (v4f*)ws;
  const int PSV = WS_PART_SP / 4;
#define LDQ(k, q) wsv[PSV + ((k) * 8 + (q)) * 8192 + gp4]
  v4f ss = LDQ(0,0) + LDQ(1,0) + LDQ(2,0) + LDQ(3,0);
  v4f ts = LDQ(0,1) + LDQ(1,1) + LDQ(2,1) + LDQ(3,1);
  v4f sx = v4max(v4max(LDQ(0,2), LDQ(1,2)), v4max(LDQ(2,2), LDQ(3,2)));
  v4f tx = v4max(v4max(LDQ(0,3), LDQ(1,3)), v4max(LDQ(2,3), LDQ(3,3)));
  v4f lo = LDQ(0,4) + LDQ(1,4) + LDQ(2,4) + LDQ(3,4);
  v4f co = LDQ(0,5) + LDQ(1,5) + LDQ(2,5) + LDQ(3,5);
  v4f lh = LDQ(0,6) + LDQ(1,6) + LDQ(2,6) + LDQ(3,6);
  v4f ch = LDQ(0,7) + LDQ(1,7) + LDQ(2,7) + LDQ(3,7);
#undef LDQ
  v4f ssp, tsp, od, hd;
  for (int j = 0; j < 4; ++j) {
    ssp[j] = sigm(ss[j] * (1.0f / 256.0f) + sx[j]);
    tsp[j] = sigm(ts[j] * (1.0f / 256.0f) + tx[j]);
    float l2 = sqrtf(lo[j] + 1e-6f);
    od[j] = (l2 + 1.0f - co[j] * (1.0f / 256.0f)) * sigm(5.0f * l2);
    float l2h = sqrtf(lh[j] + 1e-6f);
    hd[j] = (l2h + 1.0f - ch[j] * (1.0f / 256.0f)) * sigm(5.0f * l2h);
  }
  wso[WS_SSP / 4 + gp4] = ssp;
  wso[WS_TSP / 4 + gp4] = tsp;
  wso[WS_OD  / 4 + gp4] = od;
  wso[WS_HD  / 4 + gp4] = hd;
}

// ---- Pass 3: 2-class softmax KL + class-1 MSE over [8,2,256,256] ------------
__global__ void __launch_bounds__(256)
kl_kernel(const float* __restrict__ SO, const float* __restrict__ TO,
          float* __restrict__ ws) {
  __shared__ float sm[256];
  const int q4 = blockIdx.x * 256 + threadIdx.x;   // v4f position [0,131072)
  const int b  = q4 >> 14;                          // 16384 v4f per image
  const int p4 = q4 & 16383;
  const v4f* SO4 = (const v4f*)SO;
  const v4f* TO4 = (const v4f*)TO;
  const size_t base4 = (size_t)b * 32768;           // 2*65536/4
  v4f s0v = __builtin_nontemporal_load(&SO4[base4 + p4]);
  v4f s1v = __builtin_nontemporal_load(&SO4[base4 + 16384 + p4]);
  v4f t0v = __builtin_nontemporal_load(&TO4[base4 + p4]);
  v4f t1v = __builtin_nontemporal_load(&TO4[base4 + 16384 + p4]);
  float kl = 0.0f, c1 = 0.0f;
  const float invT = 0.25f;  // 1/T, T=4
  for (int j = 0; j < 4; ++j) {
    float s0 = s0v[j] * invT, s1 = s1v[j] * invT;
    float t0 = t0v[j] * invT, t1 = t1v[j] * invT;
    float msx = fmaxf(s0, s1);
    float lses = msx + __logf(__expf(s0 - msx) + __expf(s1 - msx));
    float ls0 = s0 - lses, ls1 = s1 - lses;
    float mtx = fmaxf(t0, t1);
    float lset = mtx + __logf(__expf(t0 - mtx) + __expf(t1 - mtx));
    float lt0 = t0 - lset, lt1 = t1 - lset;
    float p0 = __expf(lt0), p1 = __expf(lt1);
    kl += p0 * (lt0 - ls0) + p1 * (lt1 - ls1);
    float e1 = __expf(ls1);
    float dd = e1 - p1;
    c1 += dd * dd;
  }
  float kls = block_sum_256(kl, sm);
  float c1s = block_sum_256(c1, sm);
  if (threadIdx.x == 0) {
    ws[WS_PART_KL + blockIdx.x] = kls;
    ws[WS_PART_C1 + blockIdx.x] = c1s;
  }
}

// ---- Pass 4: single-block finalize ------------------------------------------
__global__ void __launch_bounds__(256)
finalize_kernel(const float* __restrict__ ws, float* __restrict__ out) {
  __shared__ float sm[256];
  const int tid = threadIdx.x;

  // phase 1: mean(opt_diff), needed before the mask-dependent sums
  float a = 0.0f;
  for (int i = tid; i < BHW_; i += 256) a += ws[WS_OD + i];
  float sum_od  = block_sum_256(a, sm);
  float mean_od = sum_od / 32768.0f;
  float aw  = sigm(10.0f * mean_od);
  float thr = 1.5f * mean_od;

  // phase 2: mask-weighted diff/spatial losses + channel losses + partials
  float dl = 0.0f, sl = 0.0f;
  for (int i = tid; i < BHW_; i += 256) {
    float od = ws[WS_OD + i], hd = ws[WS_HD + i];
    float mk = (od > thr) ? 1.0f : 0.0f;
    float w  = mk * 2.0f + (1.0f - mk) * 0.5f;
    float d  = (hd - od) * w;
    dl += d * d;
    float amp = 1.0f + aw * mk;
    float ds  = (ws[WS_SSP + i] - ws[WS_TSP + i]) * amp;
    sl += ds * ds;
  }
  float g = 0.0f, l = 0.0f, clf = 0.0f, cl = 0.0f;
  for (int i = tid; i < BC_; i += 256) {
    g += ws[WS_PART_G + i];
    l += ws[WS_PART_L + i];
    float msn = ws[WS_CHAN_S + i] * (1.0f / 4096.0f);
    float mtn = ws[WS_CHAN_T + i] * (1.0f / 4096.0f);
    float dcf = msn - mtn;              clf += dcf * dcf;
    float dch = sigm(msn) - sigm(mtn);  cl  += dch * dch;
  }
  float kl = 0.0f, c1 = 0.0f;
  for (int i = tid; i < KL_BLOCKS; i += 256) {
    kl += ws[WS_PART_KL + i];
    c1 += ws[WS_PART_C1 + i];
  }
  float sg   = block_sum_256(g, sm);
  float slc  = block_sum_256(l, sm);
  float sclf = block_sum_256(clf, sm);
  float scl  = block_sum_256(cl, sm);
  float sdl  = block_sum_256(dl, sm);
  float ssl  = block_sum_256(sl, sm);
  float skl  = block_sum_256(kl, sm);
  float sc1  = block_sum_256(c1, sm);

  if (tid == 0) {
    float global_loss = sg / FEAT_N;
    float local_loss  = slc / FEAT_N;
    float channel_f   = sclf / 2048.0f;
    float feat = 0.3f * global_loss + 0.5f * local_loss + 0.2f * channel_f;

    float kl_loss = (skl / 8.0f) * 16.0f;          // batchmean * T^2
    float class1  = (sc1 / 524288.0f) * 2.0f;
    float outl = kl_loss + class1;

    float diff_loss    = sdl / 32768.0f;
    float spatial_loss = ssl / 32768.0f;
    float channel_loss = scl / 2048.0f;
    float alpha = 0.5f * (1.0f + 0.5f * aw);
    float beta  = 0.3f * (1.0f - 0.3f * aw);
    float gamma = 0.2f * (1.0f + 0.5f * aw);
    float datt = alpha * diff_loss + beta * channel_loss + gamma * spatial_loss;

    float total = 0.3f * feat + 0.4f * outl + 0.3f * datt;
    out[0] = total; out[1] = feat; out[2] = outl; out[3] = datt;
  }
}

extern "C" void kernel_launch(void* const* d_in, const int* in_sizes, int n_in,
                              void* d_out, int out_size, void* d_ws, size_t ws_size,
                              hipStream_t stream) {
  (void)in_sizes; (void)n_in; (void)out_size; (void)ws_size;
  const float* S  = (const float*)d_in[0];  // student_features
  const float* T  = (const float*)d_in[1];  // teacher_features
  const float* SO = (const float*)d_in[2];  // student_outputs
  const float* TO = (const float*)d_in[3];  // teacher_outputs
  const float* O1 = (const float*)d_in[4];  // opt_t1
  const float* O2 = (const float*)d_in[5];  // opt_t2
  const float* SR = (const float*)d_in[6];  // sar_t2
  const float* M  = (const float*)d_in[7];  // feature_mask
  float* ws  = (float*)d_ws;
  float* out = (float*)d_out;

  feat_channel_kernel        <<<dim3(BC_),       dim3(256), 0, stream>>>(S, T, M, ws);
  spatial_diff_partial_kernel<<<dim3(128),       dim3(256), 0, stream>>>(S, T, O1, O2, SR, ws);
  spatial_diff_combine_kernel<<<dim3(32),        dim3(256), 0, stream>>>(ws);
  kl_kernel                  <<<dim3(KL_BLOCKS), dim3(256), 0, stream>>>(SO, TO, ws);
  finalize_kernel            <<<dim3(1),         dim3(256), 0, stream>>>(ws, out);
}